// BatchedMergedLinear_64828236366063
// MI455X (gfx1250) — compile-verified
//
#include <hip/hip_runtime.h>

typedef __attribute__((ext_vector_type(2))) float v2f;
typedef __attribute__((ext_vector_type(8))) float v8f;
typedef __attribute__((ext_vector_type(4))) int   v4i;

typedef __attribute__((address_space(1))) v4i* glb_v4i_ptr;  // global int4*
typedef __attribute__((address_space(3))) v4i* lds_v4i_ptr;  // LDS int4*

#define D_FEAT     4096
#define N_OUT      4096
#define TOK_PER_AD 2048
#define RANK       16
#define SCALING    2.0f
#define BK         32
#define LDS_STRIDE 36   // 36 floats = 144 B row pitch (16B aligned, bank-skewed)

#if __has_builtin(__builtin_amdgcn_global_load_async_to_lds_b128)
#define USE_ASYNC_LDS 1
#else
#define USE_ASYNC_LDS 0
#endif

// ---------------------------------------------------------------------------
// Kernel A: T[t][j] = SCALING * sum_d x[t][d] * loraA[m][j][d]
//   m = t / 2048, j in [0, 32).  Output T is [16384, 32] fp32 in d_ws.
//   Each wave: 16 tokens x 32 cols, K=4096 via v_wmma_f32_16x16x4_f32.
// ---------------------------------------------------------------------------
__global__ __launch_bounds__(256) void lora_proj_kernel(
    const float* __restrict__ x, const float* __restrict__ loraA,
    float* __restrict__ T)
{
  const int lane = threadIdx.x & 31;
  const int wave = threadIdx.x >> 5;
  const int hi   = lane >> 4;     // K half-select (ISA 16x4 A layout)
  const int row  = lane & 15;     // M (A) / N (B) index within tile
  const int t0   = blockIdx.x * 128 + wave * 16;
  const int m    = t0 / TOK_PER_AD;

  const float* xp = x + (size_t)(t0 + row) * D_FEAT + 2 * hi;
  const float* a0 = loraA + ((size_t)m * 32 + row) * D_FEAT + 2 * hi;
  const float* a1 = a0 + (size_t)16 * D_FEAT;

  v8f c0 = {}; v8f c1 = {};
  for (int k = 0; k < D_FEAT; k += 4) {
    v2f af = *(const v2f*)(xp + k);   // A frag: x rows, K = k+2*hi, k+2*hi+1
    v2f b0 = *(const v2f*)(a0 + k);   // B frag: loraA rows j=row (group 0)
    v2f b1 = *(const v2f*)(a1 + k);   // B frag: loraA rows j=16+row (group 1)
    c0 = __builtin_amdgcn_wmma_f32_16x16x4_f32(false, af, false, b0, (short)0, c0, false, false);
    c1 = __builtin_amdgcn_wmma_f32_16x16x4_f32(false, af, false, b1, (short)0, c1, false, false);
  }
  // C layout: VGPR v -> row (v + 8*hi), col = lane%16
  float* tb = T + (size_t)t0 * 32;
#pragma unroll
  for (int v = 0; v < 8; ++v) {
    int r = v + 8 * hi;
    tb[(size_t)r * 32 + row]      = SCALING * c0[v];
    tb[(size_t)r * 32 + 16 + row] = SCALING * c1[v];
  }
}

// ---------------------------------------------------------------------------
// Kernel B: out = X @ W^T + bias, with fused rank-16 LoRA epilogue on
// column blocks [0,1024) (g=0) and [2048,3072) (g=1).
// 128x128 tile per workgroup, 8 waves (4 along M, 2 along N),
// each wave 32x64 = 2x4 WMMA f32 16x16 accumulators.
// K staged through double-buffered LDS filled by async global->LDS copies
// (ASYNCcnt), overlapping the next chunk's fill with the current WMMAs.
// ---------------------------------------------------------------------------
__global__ __launch_bounds__(256) void merged_gemm_kernel(
    const float* __restrict__ x, const float* __restrict__ w,
    const float* __restrict__ bias, const float* __restrict__ loraB,
    const float* __restrict__ T, float* __restrict__ out)
{
  __shared__ float Xs[2][128][LDS_STRIDE];
  __shared__ float Ws[2][128][LDS_STRIDE];

  const int tid   = threadIdx.x;
  const int lane  = tid & 31;
  const int wave  = tid >> 5;
  const int hi    = lane >> 4;
  const int row   = lane & 15;
  const int t0    = blockIdx.x * 128;   // token tile base
  const int n0    = blockIdx.y * 128;   // output-column tile base
  const int waveM = wave & 3;
  const int waveN = wave >> 2;
  const int m0    = waveM * 32;
  const int nw    = waveN * 64;

  // cooperative loader indices: 256 lanes x 16B -> 32 rows/pass, 4 passes
  const int lr = tid >> 3;           // 0..31
  const int lc = (tid & 7) * 4;      // 0,4,...,28

  const float* xbase = x + (size_t)(t0 + lr) * D_FEAT + lc;
  const float* wbase = w + (size_t)(n0 + lr) * D_FEAT + lc;

  v8f c[2][4] = {};

#define ISSUE_CHUNK(K0, BUF)                                                   \
  do {                                                                         \
    _Pragma("unroll")                                                          \
    for (int p = 0; p < 4; ++p) {                                              \
      IssueCopy(xbase + (size_t)(32 * p) * D_FEAT + (K0),                      \
                &Xs[(BUF)][lr + 32 * p][lc]);                                  \
      IssueCopy(wbase + (size_t)(32 * p) * D_FEAT + (K0),                      \
                &Ws[(BUF)][lr + 32 * p][lc]);                                  \
    }                                                                          \
  } while (0)

#if USE_ASYNC_LDS
  auto IssueCopy = [](const float* g, float* l) {
    __builtin_amdgcn_global_load_async_to_lds_b128(
        (glb_v4i_ptr)g, (lds_v4i_ptr)l, 0, 0);
  };
#define WAIT_COPIES() asm volatile("s_wait_asynccnt 0x0" ::: "memory")
#else
  auto IssueCopy = [](const float* g, float* l) {
    *(float4*)l = *(const float4*)g;   // sync fallback: vmem -> vgpr -> ds
  };
#define WAIT_COPIES() ((void)0)
#endif

  // ---- pipelined main loop: one barrier per K chunk ----
  ISSUE_CHUNK(0, 0);
  WAIT_COPIES();
  __syncthreads();

  const int NCHUNK = D_FEAT / BK;    // 128
  for (int i = 0; i < NCHUNK; ++i) {
    const int cur = i & 1;
    if (i + 1 < NCHUNK) ISSUE_CHUNK((i + 1) * BK, cur ^ 1);

#pragma unroll
    for (int ks = 0; ks < BK; ks += 4) {
      v2f af[2]; v2f bf[4];
#pragma unroll
      for (int mi = 0; mi < 2; ++mi)
        af[mi] = *(const v2f*)&Xs[cur][m0 + mi * 16 + row][ks + 2 * hi];
#pragma unroll
      for (int ni = 0; ni < 4; ++ni)
        bf[ni] = *(const v2f*)&Ws[cur][nw + ni * 16 + row][ks + 2 * hi];
#pragma unroll
      for (int mi = 0; mi < 2; ++mi)
#pragma unroll
        for (int ni = 0; ni < 4; ++ni)
          c[mi][ni] = __builtin_amdgcn_wmma_f32_16x16x4_f32(
              false, af[mi], false, bf[ni], (short)0, c[mi][ni], false, false);
    }

    WAIT_COPIES();      // my chunk i+1 asyncs have landed in LDS
    __syncthreads();    // everyone done computing buf cur + everyone's fills done
  }

  // ---- fused LoRA epilogue (rank 16; scaling already folded into T) ----
  const int g = (n0 < 1024) ? 0 : ((n0 >= 2048 && n0 < 3072) ? 1 : -1);
  if (g >= 0) {                                  // uniform per block: EXEC stays full
    const int m = t0 / TOK_PER_AD;
    const int blockBase = g * 2048;
#pragma unroll
    for (int ks = 0; ks < RANK; ks += 4) {
      v2f af[2]; v2f bf[4];
#pragma unroll
      for (int mi = 0; mi < 2; ++mi)
        af[mi] = *(const v2f*)(T + (size_t)(t0 + m0 + mi * 16 + row) * 32
                               + g * RANK + ks + 2 * hi);
#pragma unroll
      for (int ni = 0; ni < 4; ++ni) {
        int n  = n0 + nw + ni * 16 + row;
        int go = g * 1024 + (n - blockBase);     // row into [G*BLOCK, R] view
        bf[ni] = *(const v2f*)(loraB + ((size_t)m * 2048 + go) * RANK + ks + 2 * hi);
      }
#pragma unroll
      for (int mi = 0; mi < 2; ++mi)
#pragma unroll
        for (int ni = 0; ni < 4; ++ni)
          c[mi][ni] = __builtin_amdgcn_wmma_f32_16x16x4_f32(
              false, af[mi], false, bf[ni], (short)0, c[mi][ni], false, false);
    }
  }

  // ---- bias add + store ----
#pragma unroll
  for (int mi = 0; mi < 2; ++mi) {
#pragma unroll
    for (int ni = 0; ni < 4; ++ni) {
      int n = n0 + nw + ni * 16 + row;
      float bv = bias[n];
#pragma unroll
      for (int v = 0; v < 8; ++v) {
        int r = t0 + m0 + mi * 16 + v + 8 * hi;
        out[(size_t)r * N_OUT + n] = c[mi][ni][v] + bv;
      }
    }
  }
#undef ISSUE_CHUNK
#undef WAIT_COPIES
}

// ---------------------------------------------------------------------------
extern "C" void kernel_launch(void* const* d_in, const int* in_sizes, int n_in,
                              void* d_out, int out_size, void* d_ws, size_t ws_size,
                              hipStream_t stream) {
  const float* x     = (const float*)d_in[0];  // [16384, 4096]
  const float* w     = (const float*)d_in[1];  // [4096, 4096]
  const float* bias  = (const float*)d_in[2];  // [4096]
  const float* loraA = (const float*)d_in[3];  // [8, 32, 4096]
  const float* loraB = (const float*)d_in[4];  // [8, 2048, 16]
  float* out = (float*)d_out;                  // [16384, 4096]
  float* T   = (float*)d_ws;                   // [16384, 32] scratch (2 MB)

  lora_proj_kernel<<<dim3(128), dim3(256), 0, stream>>>(x, loraA, T);
  merged_gemm_kernel<<<dim3(128, 32), dim3(256), 0, stream>>>(x, w, bias, loraB, T, out);
}